// GCN_11458972745815
// MI455X (gfx1250) — compile-verified
//
#include <hip/hip_runtime.h>
#include <hip/hip_bf16.h>
#include <math.h>

// ---------------------------------------------------------------------------
// GCN forward on MI455X (gfx1250, wave32).
//   N=100000 nodes, E=1600000 edges, F_IN=256, H=64, C=10
// Memory-bound overall (~1 GB of edge-gather traffic @ 23.3 TB/s); GEMMs are
// small so we use the exact-precision V_WMMA_F32_16X16X4_F32 path.
// ---------------------------------------------------------------------------

#define GCN_N     100000
#define GCN_E     1600000
#define GCN_FIN   256
#define GCN_H     64
#define GCN_C     10

typedef float v2f __attribute__((ext_vector_type(2)));
typedef float v4f __attribute__((ext_vector_type(4)));
typedef float v8f __attribute__((ext_vector_type(8)));

// Non-returning, device-scope f32 atomic add -> global_atomic_add_f32.
__device__ __forceinline__ void atomic_add_f32(float* p, float v) {
    (void)__hip_atomic_fetch_add(p, v, __ATOMIC_RELAXED, __HIP_MEMORY_SCOPE_AGENT);
}

// ---------------- init: deg = 1 (self loop), pooled = 0 --------------------
__global__ void gcn_init_kernel(float* __restrict__ deg, float* __restrict__ pooled) {
    int t = blockIdx.x * blockDim.x + threadIdx.x;
    if (t < GCN_N) deg[t] = 1.0f;
    if (t < GCN_H) pooled[t] = 0.0f;
}

// ---------------- degree accumulation --------------------------------------
__global__ void gcn_degree_kernel(const int* __restrict__ dst, float* __restrict__ deg) {
    int t = blockIdx.x * blockDim.x + threadIdx.x;
    if (t < GCN_E) atomic_add_f32(&deg[dst[t]], 1.0f);
}

// ---------------- dinv = rsqrt(deg) ----------------------------------------
__global__ void gcn_dinv_kernel(const float* __restrict__ deg, float* __restrict__ dinv) {
    int t = blockIdx.x * blockDim.x + threadIdx.x;
    if (t < GCN_N) dinv[t] = rsqrtf(deg[t]);
}

// ---------------- WMMA f32 GEMM: C[M x 64] = A[M x K] * B[K x 64] ----------
// One wave32 computes one 16x16 tile of C, stepping K by 4 with
// V_WMMA_F32_16X16X4_F32.  A-fragment: lane half selects K pair (k,k+1) vs
// (k+2,k+3) -> one contiguous float2 load.  B-fragment mirrors the K split.
__global__ __launch_bounds__(256) void gcn_gemm_wmma_kernel(
        const float* __restrict__ A, const float* __restrict__ B,
        float* __restrict__ C, int M, int K) {
    const int lane   = threadIdx.x & 31;
    const int wave   = threadIdx.x >> 5;
    const int tile   = blockIdx.x * 8 + wave;        // 8 waves per block
    const int tilesN = GCN_H / 16;                   // 4 tiles across columns
    const int tM     = tile / tilesN;
    const int tN     = tile % tilesN;
    if (tM * 16 >= M) return;                        // wave-uniform exit

    const int half  = lane >> 4;                     // 0: lanes 0-15, 1: 16-31
    const int l16   = lane & 15;
    const int rowA  = tM * 16 + l16;                 // A row held by this lane
    const int colB  = tN * 16 + l16;                 // B/C column for this lane

    const float* __restrict__ arow = A + (long long)rowA * K;

    v8f acc = {};
    for (int k = 0; k < K; k += 4) {
        const int kk = k + (half << 1);              // k or k+2
        v2f a = *(const v2f*)(arow + kk);            // K=kk, kk+1  (8B aligned)
        v2f b;
        b.x = B[(kk + 0) * GCN_H + colB];
        b.y = B[(kk + 1) * GCN_H + colB];
        acc = __builtin_amdgcn_wmma_f32_16x16x4_f32(
                  /*neg_a=*/false, a, /*neg_b=*/false, b,
                  /*c_mod=*/(short)0, acc, /*reuse_a=*/false, /*reuse_b=*/false);
    }

    // D layout: VGPR j -> row (tM*16 + j + 8*half), column colB
    const int rbase = tM * 16 + (half << 3);
#pragma unroll
    for (int j = 0; j < 8; ++j)
        C[(long long)(rbase + j) * GCN_H + colB] = acc[j];
}

// ---------------- aggregation init: self-loop term + bias ------------------
__global__ void gcn_agg_init_kernel(const float* __restrict__ h,
                                    const float* __restrict__ dinv,
                                    const float* __restrict__ bias,
                                    float* __restrict__ agg) {
    int t = blockIdx.x * blockDim.x + threadIdx.x;
    if (t >= GCN_N * GCN_H) return;
    int i = t >> 6;                  // node
    int f = t & 63;                  // feature
    float d = dinv[i];
    agg[t] = h[t] * (d * d) + bias[f];
}

// ---------------- edge scatter-add: agg[dst] += coef * h[src] --------------
// 16 threads per edge; each thread gathers a float4 of h[src] (one
// global_load_b128) and issues 4 non-returning global_atomic_add_f32.  The
// 16 threads of an edge are consecutive, so the gather is a coalesced 256B
// burst per edge and src/dst/dinv reads hit L0 for 15 of 16 lanes.
__global__ __launch_bounds__(256) void gcn_agg_edges_kernel(
        const float* __restrict__ h, const float* __restrict__ dinv,
        const int* __restrict__ src, const int* __restrict__ dst,
        float* __restrict__ agg) {
    long long t = (long long)blockIdx.x * blockDim.x + threadIdx.x;
    int e = (int)(t >> 4);
    if (e >= GCN_E) return;
    int q = ((int)t & 15) << 2;              // feature quad base: 0,4,...,60
    int s = src[e];
    int d = dst[e];
    float coef = dinv[s] * dinv[d];
    v4f hv = *(const v4f*)(h + s * GCN_H + q);   // 16B-aligned b128 load
    float* ap = agg + d * GCN_H + q;
    atomic_add_f32(ap + 0, hv.x * coef);
    atomic_add_f32(ap + 1, hv.y * coef);
    atomic_add_f32(ap + 2, hv.z * coef);
    atomic_add_f32(ap + 3, hv.w * coef);
}

// ---------------- elementwise ReLU -----------------------------------------
__global__ void gcn_relu_kernel(float* __restrict__ x) {
    int t = blockIdx.x * blockDim.x + threadIdx.x;
    if (t < GCN_N * GCN_H) x[t] = fmaxf(x[t], 0.0f);
}

// ---------------- mean pool partial reduction ------------------------------
__global__ void gcn_pool_kernel(const float* __restrict__ h,
                                float* __restrict__ pooled, int totThreads) {
    int t = blockIdx.x * blockDim.x + threadIdx.x;
    int f = t & 63;
    int r0 = t >> 6;
    int stride = totThreads >> 6;
    float s = 0.0f;
    for (int r = r0; r < GCN_N; r += stride)
        s += h[r * GCN_H + f];
    atomic_add_f32(&pooled[f], s);
}

// ---------------- head: sigmoid(mean @ Wl + bl) ----------------------------
__global__ void gcn_head_kernel(const float* __restrict__ pooled,
                                const float* __restrict__ Wl,
                                const float* __restrict__ bl,
                                float* __restrict__ out) {
    int c = threadIdx.x;
    if (c >= GCN_C) return;
    const float invN = 1.0f / (float)GCN_N;
    float s = bl[c];
    for (int hh = 0; hh < GCN_H; ++hh)
        s += (pooled[hh] * invN) * Wl[hh * GCN_C + c];
    out[c] = 1.0f / (1.0f + expf(-s));
}

// ---------------------------------------------------------------------------
extern "C" void kernel_launch(void* const* d_in, const int* in_sizes, int n_in,
                              void* d_out, int out_size, void* d_ws, size_t ws_size,
                              hipStream_t stream) {
    const float* x   = (const float*)d_in[0];          // [N, 256]
    const int*   ei  = (const int*)d_in[1];            // [2, E]
    const float* W1  = (const float*)d_in[2];          // [256, 64]
    const float* b1  = (const float*)d_in[3];          // [64]
    const float* W2  = (const float*)d_in[4];          // [64, 64]
    const float* b2  = (const float*)d_in[5];          // [64]
    const float* Wl  = (const float*)d_in[6];          // [64, 10]
    const float* bl  = (const float*)d_in[7];          // [10]
    float* out = (float*)d_out;                        // [1, 10]

    const int* src = ei;                               // first E ints
    const int* dst = ei + GCN_E;                       // second E ints

    // Workspace layout (floats)
    float* ws     = (float*)d_ws;
    float* deg    = ws;                                // N
    float* dinv   = ws + GCN_N;                        // N
    float* bufA   = ws + 2 * GCN_N;                    // N*64 (h = X@W)
    float* bufB   = bufA + (size_t)GCN_N * GCN_H;      // N*64 (aggregated)
    float* pooled = bufB + (size_t)GCN_N * GCN_H;      // 64

    const int nh = GCN_N * GCN_H;                      // 6.4M elements
    const int gemmBlocks = (GCN_N / 16) * (GCN_H / 16) / 8;   // 3125
    const long long edgeThreads = (long long)GCN_E * 16;      // 25.6M
    const int edgeBlocks = (int)((edgeThreads + 255) / 256);

    // 1) init deg (self-loop) and pooled accumulator
    gcn_init_kernel<<<(GCN_N + 255) / 256, 256, 0, stream>>>(deg, pooled);
    // 2) degrees
    gcn_degree_kernel<<<(GCN_E + 255) / 256, 256, 0, stream>>>(dst, deg);
    // 3) dinv
    gcn_dinv_kernel<<<(GCN_N + 255) / 256, 256, 0, stream>>>(deg, dinv);

    // ---- layer 1 ----
    gcn_gemm_wmma_kernel<<<gemmBlocks, 256, 0, stream>>>(x, W1, bufA, GCN_N, GCN_FIN);
    gcn_agg_init_kernel<<<(nh + 255) / 256, 256, 0, stream>>>(bufA, dinv, b1, bufB);
    gcn_agg_edges_kernel<<<edgeBlocks, 256, 0, stream>>>(bufA, dinv, src, dst, bufB);
    gcn_relu_kernel<<<(nh + 255) / 256, 256, 0, stream>>>(bufB);

    // ---- layer 2 ----
    gcn_gemm_wmma_kernel<<<gemmBlocks, 256, 0, stream>>>(bufB, W2, bufA, GCN_N, GCN_H);
    gcn_agg_init_kernel<<<(nh + 255) / 256, 256, 0, stream>>>(bufA, dinv, b2, bufB);
    gcn_agg_edges_kernel<<<edgeBlocks, 256, 0, stream>>>(bufA, dinv, src, dst, bufB);

    // ---- pool + head ----
    const int poolThreads = 1024 * 256;                // 4096-row stride per column
    gcn_pool_kernel<<<1024, 256, 0, stream>>>(bufB, pooled, poolThreads);
    gcn_head_kernel<<<1, 32, 0, stream>>>(pooled, Wl, bl, out);
}